// EdgeConv1d_80358838108754
// MI455X (gfx1250) — compile-verified
//
#include <hip/hip_runtime.h>
#include <stddef.h>

#define N_NODES   100000
#define K_NB      9
#define SENTINEL  1000000.0f
#define NTILES    6251            // ceil((N+1)/16)
#define NPAD      (NTILES * 16)   // 100016 padded rows for A/B buffers

// Interleaved weight layout: 32 K-pairs, each row = 128 cols * 2 + 32 pad floats.
// lw[p*288 + col*2 + j] = Wc[K = 2p+j][col]  (8B-aligned float2 per fragment;
// 288-float stride => lanes 0-15 (pair 2kk) and 16-31 (pair 2kk+1) hit
// disjoint 32-bank halves for conflict-free ds_load_b64 across wave32).
#define WPAIR_STRIDE 288
#define WCT_FLOATS   (32 * WPAIR_STRIDE)

typedef __attribute__((ext_vector_type(2))) float v2f;
typedef __attribute__((ext_vector_type(4))) float v4f;
typedef __attribute__((ext_vector_type(8))) float v8f;

// Workspace layout (float offsets)
#define WS_WCT   0                      // WCT_FLOATS: interleaved combined weights
#define WS_BIAS  WCT_FLOATS             // 128 floats: [b, 0]
#define WS_A     (65536 / 4)            // NPAD*64 floats: A = x(W1-W2)^T + b
#define WS_B     (WS_A + NPAD * 64)     // NPAD*64 floats: B = s_pts * W2^T

// ---------------------------------------------------------------------------
// Kernel 0: build interleaved combined weight and bias.
//   value(K, o) = o<64 ? W[o,K] - W[o,64+K] : W[o-64, 64+K]
//   stored at ws[WS_WCT + (K>>1)*288 + o*2 + (K&1)]
// ---------------------------------------------------------------------------
__global__ void prep_weights(const float* __restrict__ W,
                             const float* __restrict__ b,
                             float* __restrict__ ws) {
    int i = blockIdx.x * 256 + threadIdx.x;
    for (int t = i; t < 64 * 128; t += gridDim.x * 256) {
        int K = t >> 7;                  // input channel 0..63
        int o = t & 127;                 // output column 0..127
        float v;
        if (o < 64) v = W[o * 128 + K] - W[o * 128 + 64 + K];
        else        v = W[(o - 64) * 128 + 64 + K];
        ws[WS_WCT + (K >> 1) * WPAIR_STRIDE + o * 2 + (K & 1)] = v;
    }
    if (i < 128) ws[WS_BIAS + i] = (i < 64) ? b[i] : 0.0f;
}

// ---------------------------------------------------------------------------
// Kernel 1: fused GEMM  [A | B] = s_pts (NPAD x 64) * Wc^T (64 x 128) + [b|0]
// One wave per 16-row tile; 8 waves per block share the weight LDS.
// V_WMMA_F32_16X16X4_F32 (wave32), 16 K-steps per 16x16 output tile.
// ---------------------------------------------------------------------------
__global__ void __launch_bounds__(256) gemm_AB(const float* __restrict__ x,
                                               float* ws) {
    __shared__ float lw[WCT_FLOATS];
    __shared__ float lb[128];

    int tid = threadIdx.x;
    for (int t = tid; t < WCT_FLOATS; t += 256) lw[t] = ws[WS_WCT + t];
    if (tid < 128) lb[tid] = ws[WS_BIAS + tid];
    __syncthreads();

    int wave    = tid >> 5;
    int lane    = tid & 31;
    int rowTile = blockIdx.x * 8 + wave;
    if (rowTile >= NTILES) return;           // whole-wave exit (EXEC stays full)

    int half = lane >> 4;                    // 0: K pair {0,1}, 1: K pair {2,3}
    int m    = lane & 15;                    // row within tile (A) / col (B,C)
    int r0   = rowTile * 16;
    int row  = r0 + m;

    // A fragments: per lane, float2 of x[row, kk*4 + half*2 + {0,1}]
    v2f a[16];
    if (row < N_NODES) {
        const float* xr = x + (size_t)row * 64;
        #pragma unroll
        for (int kk = 0; kk < 16; ++kk) {
            a[kk] = *(const v2f*)(xr + kk * 4 + half * 2);
        }
    } else {
        float v = (row == N_NODES) ? SENTINEL : 0.0f;   // sentinel row / padding
        #pragma unroll
        for (int kk = 0; kk < 16; ++kk) { a[kk].x = v; a[kk].y = v; }
    }

    float* Abuf = ws + WS_A;
    float* Bbuf = ws + WS_B;

    #pragma unroll
    for (int ct = 0; ct < 8; ++ct) {         // 8 output column tiles (128 cols)
        int col = ct * 16 + m;
        const float* lwc = lw + (size_t)col * 2 + half * WPAIR_STRIDE;
        v8f c = {};
        #pragma unroll
        for (int kk = 0; kk < 16; ++kk) {
            // single contiguous ds_load_b64 per fragment, even-pair friendly
            v2f bf = *(const v2f*)(lwc + (kk * 2) * WPAIR_STRIDE);
            c = __builtin_amdgcn_wmma_f32_16x16x4_f32(
                    false, a[kk], false, bf, (short)0, c, false, false);
        }
        float bias = lb[col];
        float* dst = (ct < 4) ? Abuf : Bbuf;
        int   ocol = (ct < 4) ? col : (col - 64);
        #pragma unroll
        for (int i = 0; i < 8; ++i) {        // C layout: VGPR i -> M = i + half*8
            int rr = r0 + i + half * 8;
            dst[(size_t)rr * 64 + ocol] = c[i] + bias;
        }
    }
}

// ---------------------------------------------------------------------------
// Kernel 2: out[n,:] = relu( max_k ( A[n,:] + B[idx[n,k],:] ) )
// 16 lanes x float4 per node row; B (25.6MB) is L2-resident on MI455X (192MB).
// ---------------------------------------------------------------------------
__global__ void __launch_bounds__(256) gather_max(const float* __restrict__ ws,
                                                  const long long* __restrict__ eidx,
                                                  float* __restrict__ out) {
    int gid = blockIdx.x * 256 + threadIdx.x;
    int n   = gid >> 4;
    if (n >= N_NODES) return;
    int cg  = (gid & 15) << 2;

    const float* Abuf = ws + WS_A;
    const float* Bbuf = ws + WS_B;

    v4f a  = *(const v4f*)(Abuf + (size_t)n * 64 + cg);
    v4f mx = { -3.0e38f, -3.0e38f, -3.0e38f, -3.0e38f };

    const long long* ep = eidx + (size_t)n * K_NB;   // edge_index[0]
    #pragma unroll
    for (int k = 0; k < K_NB; ++k) {
        int j = (int)ep[k];                          // in [0, N]; N = sentinel row
        v4f bv = *(const v4f*)(Bbuf + (size_t)j * 64 + cg);
        v4f h;
        h.x = a.x + bv.x; h.y = a.y + bv.y; h.z = a.z + bv.z; h.w = a.w + bv.w;
        mx.x = fmaxf(mx.x, h.x); mx.y = fmaxf(mx.y, h.y);
        mx.z = fmaxf(mx.z, h.z); mx.w = fmaxf(mx.w, h.w);
    }
    v4f r;                                           // relu after max (monotone)
    r.x = fmaxf(mx.x, 0.0f); r.y = fmaxf(mx.y, 0.0f);
    r.z = fmaxf(mx.z, 0.0f); r.w = fmaxf(mx.w, 0.0f);
    *(v4f*)(out + (size_t)n * 64 + cg) = r;
}

// ---------------------------------------------------------------------------
extern "C" void kernel_launch(void* const* d_in, const int* in_sizes, int n_in,
                              void* d_out, int out_size, void* d_ws, size_t ws_size,
                              hipStream_t stream) {
    const float*     x    = (const float*)d_in[0];
    const long long* eidx = (const long long*)d_in[1];   // int64 (2,N,K); [0] = first N*K
    const float*     W    = (const float*)d_in[2];
    const float*     b    = (const float*)d_in[3];
    float*           out  = (float*)d_out;
    float*           ws   = (float*)d_ws;

    prep_weights<<<32, 256, 0, stream>>>(W, b, ws);

    int gemmBlocks = (NTILES + 7) / 8;                   // 8 waves/block, 1 tile/wave
    gemm_AB<<<gemmBlocks, 256, 0, stream>>>(x, ws);

    int gatherBlocks = (N_NODES * 16 + 255) / 256;
    gather_max<<<gatherBlocks, 256, 0, stream>>>(ws, eidx, out);
}